// piecewise_maxpool_layer_57312043598527
// MI455X (gfx1250) — compile-verified
//
#include <hip/hip_runtime.h>
#include <stdint.h>

#define SEQ       256
#define NFILT     256
#define TILE_S    16
#define NTILES    (SEQ / TILE_S)                 // 16
#define TILE_BYTES (TILE_S * NFILT * 4)          // 16384 B
#define THREADS   256
#define LOADS_PER_TILE 4                         // 4 x b128 per thread per tile

// Async copy of 16 bytes from global address sbase plus gofs into LDS at lds_addr, tracked on ASYNCcnt.
// GVS mode: saddr holds the 64-bit scalar base, vaddr the 32-bit per-lane byte offset,
// vdst VGPR holds the LDS byte address. Input is streamed once (1 GiB >> 192 MB
// L2), so mark the loads non-temporal.
__device__ __forceinline__ void async_ld_b128_nt(unsigned lds_addr, unsigned gofs,
                                                 const void* sbase) {
  asm volatile("global_load_async_to_lds_b128 %0, %1, %2 th:TH_LOAD_NT"
               :
               : "v"(lds_addr), "v"(gofs), "s"(sbase)
               : "memory");
}

__device__ __forceinline__ void wait_async_le4() {
  asm volatile("s_wait_asynccnt 0x4" ::: "memory");
}
__device__ __forceinline__ void wait_async_le0() {
  asm volatile("s_wait_asynccnt 0x0" ::: "memory");
}

__global__ __launch_bounds__(THREADS)
void piecewise_maxpool_kernel(const float* __restrict__ conv,
                              const int* __restrict__ e1v,
                              const int* __restrict__ e2v,
                              float* __restrict__ out) {
  __shared__ __align__(16) float lds[2][TILE_S * NFILT];

  const int b   = blockIdx.x;
  const int tid = threadIdx.x;

  // Block-uniform cut points -> scalar loads, scalar branches below.
  const int e1 = e1v[b];
  const int e2 = e2v[b];

  const float* gbase = conv + (size_t)b * SEQ * NFILT;   // uniform SGPR base

  // Generic LDS pointers: the low 32 bits of the flat address are the LDS byte offset.
  const unsigned ldsBuf[2] = {
      (unsigned)(uintptr_t)&lds[0][0],
      (unsigned)(uintptr_t)&lds[1][0]
  };

  float m0 = -__builtin_inff();
  float m1 = -__builtin_inff();
  float m2 = -__builtin_inff();

  // Prefetch tile 0.
  {
    const unsigned base = ldsBuf[0];
#pragma unroll
    for (int i = 0; i < LOADS_PER_TILE; ++i) {
      const unsigned ofs = (unsigned)((i * THREADS + tid) * 16);
      async_ld_b128_nt(base + ofs, ofs, gbase);
    }
  }

  for (int t = 0; t < NTILES; ++t) {
    const int cur = t & 1;

    if (t + 1 < NTILES) {
      // Issue next tile into the other buffer, then wait for the current
      // tile's 4 loads (async loads complete in order -> asynccnt <= 4).
      const unsigned base  = ldsBuf[cur ^ 1];
      const unsigned gtile = (unsigned)((t + 1) * TILE_BYTES);
#pragma unroll
      for (int i = 0; i < LOADS_PER_TILE; ++i) {
        const unsigned ofs = (unsigned)((i * THREADS + tid) * 16);
        async_ld_b128_nt(base + ofs, gtile + ofs, gbase);
      }
      wait_async_le4();
    } else {
      wait_async_le0();
    }
    __syncthreads();                 // all waves' tile-t data visible in LDS

    const float* buf  = lds[cur];
    const int    sbas = t * TILE_S;
#pragma unroll
    for (int r = 0; r < TILE_S; ++r) {
      const float v = buf[r * NFILT + tid];   // bank-conflict-free
      const int   s = sbas + r;
      if (s <= e1)      m0 = fmaxf(m0, v);    // scalar-uniform branch
      else if (s <= e2) m1 = fmaxf(m1, v);
      else              m2 = fmaxf(m2, v);
    }
    __syncthreads();                 // protect buffer before it is refilled
  }

  // Output is write-once, never re-read: non-temporal stores.
  float* ob = out + (size_t)b * (3 * NFILT);
  __builtin_nontemporal_store(m0, ob + tid);
  __builtin_nontemporal_store(m1, ob + NFILT + tid);
  __builtin_nontemporal_store(m2, ob + 2 * NFILT + tid);
}

extern "C" void kernel_launch(void* const* d_in, const int* in_sizes, int n_in,
                              void* d_out, int out_size, void* d_ws, size_t ws_size,
                              hipStream_t stream) {
  const float* conv = (const float*)d_in[0];
  const int*   e1   = (const int*)d_in[1];
  const int*   e2   = (const int*)d_in[2];
  float*       out  = (float*)d_out;

  const int batch = in_sizes[1];   // element count of e1 == BATCH
  piecewise_maxpool_kernel<<<batch, THREADS, 0, stream>>>(conv, e1, e2, out);
}